// MiaoMiaoBlock_52003464020805
// MI455X (gfx1250) — compile-verified
//
#include <hip/hip_runtime.h>
#include <hip/hip_bf16.h>
#include <math.h>

#define CCH 32          // channels
#define NPATH 11
#define NB 8

typedef __attribute__((ext_vector_type(2))) float v2f;
typedef __attribute__((ext_vector_type(8))) float v8f;

// ---------------------------------------------------------------------------
// D = A(16x4 f32) x B(4x16 f32) + C(16x16 f32)   -- CDNA5 WMMA, wave32
// ---------------------------------------------------------------------------
__device__ __forceinline__ v8f wmma4(v2f a, v2f b, v8f c) {
  return __builtin_amdgcn_wmma_f32_16x16x4_f32(
      /*neg_a=*/false, a, /*neg_b=*/false, b,
      /*c_mod=*/(short)0, c, /*reuse_a=*/false, /*reuse_b=*/false);
}

// ---------------------------------------------------------------------------
// Kernel 0: zero the accumulator workspace (a0|a1|a2), float4 stores
// ---------------------------------------------------------------------------
__global__ void mmb_zero(float4* __restrict__ p, int n4) {
  int i = blockIdx.x * blockDim.x + threadIdx.x;
  if (i < n4) p[i] = make_float4(0.f, 0.f, 0.f, 0.f);
}

// ---------------------------------------------------------------------------
// Kernel 1: edge phase. One wave32 per edge, lane = channel.
// All rhat⊗rhat contractions collapsed to scalar dots; 1/16 folded in.
// ---------------------------------------------------------------------------
__global__ void mmb_edge(const float* __restrict__ rij,
                         const float* __restrict__ Wrad,
                         const int*   __restrict__ idx_i,
                         const int*   __restrict__ idx_j,
                         const float* __restrict__ node0,
                         const float* __restrict__ node1,
                         const float* __restrict__ node2,
                         float* __restrict__ a0,
                         float* __restrict__ a1,
                         float* __restrict__ a2,
                         int E) {
  const int e = blockIdx.x * (blockDim.x >> 5) + (threadIdx.x >> 5);
  const int c = threadIdx.x & 31;
  if (e >= E) return;

  const float r0 = rij[e * 3 + 0], r1 = rij[e * 3 + 1], r2 = rij[e * 3 + 2];
  const float d  = sqrtf(r0 * r0 + r1 * r1 + r2 * r2 + 1e-12f);
  const float iv = 1.0f / d;
  const float rh[3] = {r0 * iv, r1 * iv, r2 * iv};

  float x = d * 0.2f;                       // d / CUTOFF
  x = (x > 1.0f) ? 1.0f : x;
  const float fc = 0.5f * (cosf(3.14159265358979f * x) + 1.0f);

  float rbf[NB];
#pragma unroll
  for (int b = 0; b < NB; ++b) {
    const float t = d - (0.5f + 0.6428571429f * (float)b);   // centers
    rbf[b] = __expf(-4.0f * t * t) * fc;
  }

  float w[NPATH];
#pragma unroll
  for (int p = 0; p < NPATH; ++p) {
    float acc = 0.f;
#pragma unroll
    for (int b = 0; b < NB; ++b) acc += rbf[b] * Wrad[(p * NB + b) * CCH + c];
    w[p] = acc;
  }

  const int j = idx_j[e];
  const int i = idx_i[e];

  const float g0 = node0[j * CCH + c];
  float g1[3], g2[9];
#pragma unroll
  for (int a = 0; a < 3; ++a) g1[a] = node1[(j * CCH + c) * 3 + a];
#pragma unroll
  for (int q = 0; q < 9; ++q) g2[q] = node2[(j * CCH + c) * 9 + q];

  const float d1 = g1[0] * rh[0] + g1[1] * rh[1] + g1[2] * rh[2];
  float gr[3];
#pragma unroll
  for (int a = 0; a < 3; ++a)
    gr[a] = g2[a * 3 + 0] * rh[0] + g2[a * 3 + 1] * rh[1] + g2[a * 3 + 2] * rh[2];
  const float d2 = gr[0] * rh[0] + gr[1] * rh[1] + gr[2] * rh[2];

  const float inv16 = 0.0625f;  // fold 1/NORM_FACTOR into the messages

  atomicAdd(&a0[i * CCH + c], (g0 * w[0] + d1 * w[4] + d2 * w[9]) * inv16);

#pragma unroll
  for (int a = 0; a < 3; ++a) {
    const float m = (g0 * w[1] + d1 * w[6]) * rh[a] + g1[a] * w[3] + gr[a] * w[8];
    atomicAdd(&a1[(i * CCH + c) * 3 + a], m * inv16);
  }
#pragma unroll
  for (int a = 0; a < 3; ++a) {
    const float ca = g0 * w[2] * rh[a] + g1[a] * w[5] + gr[a] * w[10];
#pragma unroll
    for (int b = 0; b < 3; ++b) {
      const float m = ca * rh[b] + g2[a * 3 + b] * w[7];
      atomicAdd(&a2[(i * CCH + c) * 9 + a * 3 + b], m * inv16);
    }
  }
}

// ---------------------------------------------------------------------------
// Kernel 2: node phase, all 32x32 matmuls via V_WMMA_F32_16X16X4_F32.
// 4 waves / block, one 16-node tile per wave. 14 weight matrices staged
// into LDS pre-swizzled into WMMA B-operand layout.
// ---------------------------------------------------------------------------
#define W_U0 0
#define W_U1 1
#define W_U2 2
#define W_V(k) (3 + (k))
#define W_G(k) (11 + (k))

__global__ void __launch_bounds__(128)
mmb_node(const float* __restrict__ a0, const float* __restrict__ a1,
         const float* __restrict__ a2,
         const float* __restrict__ node0, const float* __restrict__ node1,
         const float* __restrict__ node2,
         const float* __restrict__ U, const float* __restrict__ V,
         const float* __restrict__ Wg,
         float* __restrict__ out0, float* __restrict__ out1,
         float* __restrict__ out2, int N) {
  // B-operand swizzled weights: [mat][kstep][coltile][lane] (float2 per lane)
  __shared__ float2 wlds[14 * 8 * 2 * 32];
  __shared__ float  ylds[4][16][CCH];   // y0 staging, per wave

  // ---- stage weights into B-operand layout ----
  for (int idx = threadIdx.x; idx < 14 * 1024; idx += blockDim.x) {
    const int wmat = idx >> 10;
    const int rem  = idx & 1023;
    const int k    = rem >> 5;     // input channel (K)
    const int n    = rem & 31;     // output channel (N)
    const float* src = (wmat < 3) ? (U + wmat * 1024)
                     : (wmat < 11) ? (V + (wmat - 3) * 1024)
                                   : (Wg + (wmat - 11) * 1024);
    const float val = src[rem];
    const int s   = k >> 2;
    const int kk  = k & 3;
    const int ln  = ((kk >> 1) << 4) | (n & 15);
    const int t   = n >> 4;
    const int jj  = kk & 1;
    ((float*)&wlds[((wmat * 8 + s) * 2 + t) * 32 + ln])[jj] = val;
  }
  __syncthreads();

  const int lane = threadIdx.x & 31;
  const int wv   = threadIdx.x >> 5;
  const int half = lane >> 4;          // K sub-slot for A/B operands
  const int m    = lane & 15;          // row (A) / column (B,D) index
  const int tile = blockIdx.x * 4 + wv;
  if (tile * 16 >= N) return;

#define LDB(wm, s, t)                                                       \
  ({ float2 _bv = wlds[(((wm) * 8 + (s)) * 2 + (t)) * 32 + lane];           \
     v2f _b; _b.x = _bv.x; _b.y = _bv.y; _b; })

#define MM2(acc, aop, wm, s)                                                \
  do { v2f _a = (aop);                                                      \
       acc[0] = wmma4(_a, LDB(wm, s, 0), acc[0]);                           \
       acc[1] = wmma4(_a, LDB(wm, s, 1), acc[1]); } while (0)

  const v8f vzero = {};

  // per-lane base loads for k-step s: channels (k, k+1), row = tile*16 + m
  const int rowA = tile * 16 + m;
  const int rowC = (rowA < N) ? rowA : (N - 1);   // clamp loads for tails
  float x0[2], b1[2][3], b2[2][9];
#define LOAD_BASE(s)                                                        \
  do { const int _k = (s) * 4 + 2 * half;                                   \
       const int _rc = rowC * CCH + _k;                                     \
       x0[0] = a0[_rc]; x0[1] = a0[_rc + 1];                                \
       _Pragma("unroll") for (int _a = 0; _a < 3; ++_a) {                   \
         b1[0][_a] = a1[_rc * 3 + _a]; b1[1][_a] = a1[(_rc + 1) * 3 + _a]; }\
       _Pragma("unroll") for (int _q = 0; _q < 9; ++_q) {                   \
         b2[0][_q] = a2[_rc * 9 + _q]; b2[1][_q] = a2[(_rc + 1) * 9 + _q]; }\
  } while (0)

  // ---------------- Pass A: y0 = a0@U0 + a0²@V0 + (Σa1²)@V3 + (Σa2²)@V6 --
  v8f y0acc[2] = {vzero, vzero};
  for (int s = 0; s < 8; ++s) {
    LOAD_BASE(s);
    v2f ax, asq, s1v, s2v;
#pragma unroll
    for (int jj = 0; jj < 2; ++jj) {
      ax[jj]  = x0[jj];
      asq[jj] = x0[jj] * x0[jj];
      float acc = 0.f;
#pragma unroll
      for (int a = 0; a < 3; ++a) acc += b1[jj][a] * b1[jj][a];
      s1v[jj] = acc;
      acc = 0.f;
#pragma unroll
      for (int q = 0; q < 9; ++q) acc += b2[jj][q] * b2[jj][q];
      s2v[jj] = acc;
    }
    MM2(y0acc, ax,  W_U0,  s);
    MM2(y0acc, asq, W_V(0), s);
    MM2(y0acc, s1v, W_V(3), s);
    MM2(y0acc, s2v, W_V(6), s);
  }

  // stage y0 tile to LDS (plain [16][32]) so it can be re-fed as A operand
#pragma unroll
  for (int t = 0; t < 2; ++t)
#pragma unroll
    for (int r = 0; r < 8; ++r)
      ylds[wv][r + 8 * half][t * 16 + m] = y0acc[t][r];
  asm volatile("s_wait_dscnt 0" ::: "memory");   // intra-wave LDS RAW fence

  // ---------------- gates: silu(y0 @ Wg{0,1,2}) --------------------------
  v8f ga[3][2] = {{vzero, vzero}, {vzero, vzero}, {vzero, vzero}};
  for (int s = 0; s < 8; ++s) {
    const int k = s * 4 + 2 * half;
    v2f ay; ay.x = ylds[wv][m][k]; ay.y = ylds[wv][m][k + 1];
#pragma unroll
    for (int g = 0; g < 3; ++g) MM2(ga[g], ay, W_G(g), s);
  }
  v8f gate[3][2];
#pragma unroll
  for (int g = 0; g < 3; ++g)
#pragma unroll
    for (int t = 0; t < 2; ++t)
#pragma unroll
      for (int r = 0; r < 8; ++r) {
        const float v = ga[g][t][r];
        gate[g][t][r] = v * (1.0f / (1.0f + __expf(-v)));
      }

  // new0 = node0 + silu(y0@Wg0)
#pragma unroll
  for (int t = 0; t < 2; ++t)
#pragma unroll
    for (int r = 0; r < 8; ++r) {
      const int row = tile * 16 + r + 8 * half;
      if (row < N) {
        const int cc = t * 16 + m;
        out0[row * CCH + cc] = node0[row * CCH + cc] + gate[0][t][r];
      }
    }

  // ---------------- Pass B: y1 (3 spatial comps) -------------------------
  v8f y1acc[3][2] = {{vzero, vzero}, {vzero, vzero}, {vzero, vzero}};
  for (int s = 0; s < 8; ++s) {
    LOAD_BASE(s);
#pragma unroll
    for (int v = 0; v < 3; ++v) {
      v2f av, a01, tv;
#pragma unroll
      for (int jj = 0; jj < 2; ++jj) {
        av[jj]  = b1[jj][v];
        a01[jj] = x0[jj] * b1[jj][v];
        float acc = 0.f;
#pragma unroll
        for (int a = 0; a < 3; ++a) acc += b1[jj][a] * b2[jj][a * 3 + v];
        tv[jj] = acc;
      }
      MM2(y1acc[v], av,  W_U1,  s);
      MM2(y1acc[v], a01, W_V(1), s);
      MM2(y1acc[v], tv,  W_V(5), s);
    }
  }
#pragma unroll
  for (int v = 0; v < 3; ++v)
#pragma unroll
    for (int t = 0; t < 2; ++t)
#pragma unroll
      for (int r = 0; r < 8; ++r) {
        const int row = tile * 16 + r + 8 * half;
        if (row < N) {
          const int cc = t * 16 + m;
          const int o  = (row * CCH + cc) * 3 + v;
          out1[o] = node1[o] + y1acc[v][t][r] * gate[1][t][r];
        }
      }

  // ---------------- Pass C: y2, 3 sub-passes over spatial row 'a' --------
  for (int aIdx = 0; aIdx < 3; ++aIdx) {
    v8f y2acc[3][2] = {{vzero, vzero}, {vzero, vzero}, {vzero, vzero}};
    for (int s = 0; s < 8; ++s) {
      LOAD_BASE(s);
#pragma unroll
      for (int b = 0; b < 3; ++b) {
        v2f a2v, a02, a11, qv;
#pragma unroll
        for (int jj = 0; jj < 2; ++jj) {
          const float g = b2[jj][aIdx * 3 + b];
          a2v[jj] = g;
          a02[jj] = x0[jj] * g;
          a11[jj] = b1[jj][aIdx] * b1[jj][b];
          float acc = 0.f;
#pragma unroll
          for (int kk = 0; kk < 3; ++kk)
            acc += b2[jj][aIdx * 3 + kk] * b2[jj][kk * 3 + b];
          qv[jj] = acc;
        }
        MM2(y2acc[b], a2v, W_U2,  s);
        MM2(y2acc[b], a02, W_V(2), s);
        MM2(y2acc[b], a11, W_V(4), s);
        MM2(y2acc[b], qv,  W_V(7), s);
      }
    }
#pragma unroll
    for (int b = 0; b < 3; ++b)
#pragma unroll
      for (int t = 0; t < 2; ++t)
#pragma unroll
        for (int r = 0; r < 8; ++r) {
          const int row = tile * 16 + r + 8 * half;
          if (row < N) {
            const int cc = t * 16 + m;
            const int o  = (row * CCH + cc) * 9 + aIdx * 3 + b;
            out2[o] = node2[o] + y2acc[b][t][r] * gate[2][t][r];
          }
        }
  }
#undef LOAD_BASE
#undef MM2
#undef LDB
}

// ---------------------------------------------------------------------------
extern "C" void kernel_launch(void* const* d_in, const int* in_sizes, int n_in,
                              void* d_out, int out_size, void* d_ws, size_t ws_size,
                              hipStream_t stream) {
  const float* node0 = (const float*)d_in[0];
  const float* node1 = (const float*)d_in[1];
  const float* node2 = (const float*)d_in[2];
  const float* rij   = (const float*)d_in[3];
  const float* Wrad  = (const float*)d_in[4];
  const float* U     = (const float*)d_in[5];
  const float* V     = (const float*)d_in[6];
  const float* Wg    = (const float*)d_in[7];
  const int*   idx_i = (const int*)d_in[8];
  const int*   idx_j = (const int*)d_in[9];

  const int N = in_sizes[0] / CCH;
  const int E = in_sizes[3] / 3;

  float* a0 = (float*)d_ws;                   // [N,32]
  float* a1 = a0 + (size_t)N * CCH;           // [N,32,3]
  float* a2 = a1 + (size_t)N * CCH * 3;       // [N,32,9]

  float* out0 = (float*)d_out;
  float* out1 = out0 + (size_t)N * CCH;
  float* out2 = out1 + (size_t)N * CCH * 3;

  // zero the accumulator workspace (13*N*32 floats, divisible by 4)
  const int n4 = N * CCH * 13 / 4;
  mmb_zero<<<(n4 + 255) / 256, 256, 0, stream>>>((float4*)d_ws, n4);

  // edge scatter: one wave per edge, 8 waves / block
  mmb_edge<<<(E + 7) / 8, 256, 0, stream>>>(rij, Wrad, idx_i, idx_j,
                                            node0, node1, node2,
                                            a0, a1, a2, E);

  // node phase: 4 wave-tiles (64 nodes) per block
  const int nTiles = (N + 15) / 16;
  mmb_node<<<(nTiles + 3) / 4, 128, 0, stream>>>(a0, a1, a2,
                                                 node0, node1, node2,
                                                 U, V, Wg,
                                                 out0, out1, out2, N);
}